// DeepSignatureModel_35089882808697
// MI455X (gfx1250) — compile-verified
//
#include <hip/hip_runtime.h>

typedef float v2f __attribute__((ext_vector_type(2)));
typedef float v8f __attribute__((ext_vector_type(8)));

#define B_N  64
#define L_N  1024
#define L1   1021      // stream length after augment1 (K=4)
#define C1   8
#define SIG1 584       // 8 + 64 + 512
#define L2   1018      // stream length after augment2 (K=4)

static constexpr size_t A_SZ = (size_t)B_N * L1 * C1;     // floats
static constexpr size_t S_SZ = (size_t)B_N * L1 * SIG1;   // floats (152.6 MB, fits L2)

// ---------------------------------------------------------------------------
// K1: augment1. a[b,t,:] = [x[b,t+3,0..2], t/1020, conv(x)[t,0..3]]
// ---------------------------------------------------------------------------
__global__ __launch_bounds__(256) void augment1_kernel(
    const float* __restrict__ x,
    const float* __restrict__ w0, const float* __restrict__ b0,
    const float* __restrict__ w1, const float* __restrict__ b1,
    const float* __restrict__ w2, const float* __restrict__ b2,
    float* __restrict__ a) {
  int gid = blockIdx.x * 256 + threadIdx.x;
  if (gid >= B_N * L1) return;
  int batch = gid / L1;
  int t = gid - batch * L1;
  const float* xr = x + ((size_t)batch * L_N + t) * 3;

  float xin[12];
#pragma unroll
  for (int i = 0; i < 12; ++i) xin[i] = xr[i];   // x[b,t..t+3,0..2] contiguous

  float y0[8];
#pragma unroll
  for (int n = 0; n < 8; ++n) y0[n] = b0[n];
#pragma unroll
  for (int i = 0; i < 12; ++i)
#pragma unroll
    for (int n = 0; n < 8; ++n) y0[n] += xin[i] * w0[i * 8 + n];
#pragma unroll
  for (int n = 0; n < 8; ++n) y0[n] = y0[n] > 0.f ? y0[n] : 0.f;

  float y1[8];
#pragma unroll
  for (int n = 0; n < 8; ++n) {
    float s = b1[n];
#pragma unroll
    for (int c = 0; c < 8; ++c) s += y0[c] * w1[c * 8 + n];
    y1[n] = s > 0.f ? s : 0.f;
  }
  float y2[4];
#pragma unroll
  for (int n = 0; n < 4; ++n) {
    float s = b2[n];
#pragma unroll
    for (int c = 0; c < 8; ++c) s += y1[c] * w2[c * 4 + n];
    y2[n] = s;
  }
  float* ar = a + (size_t)gid * C1;
  ar[0] = xr[9]; ar[1] = xr[10]; ar[2] = xr[11];   // x[b,t+3,:]
  ar[3] = (float)t * (1.0f / 1020.0f);             // time channel
  ar[4] = y2[0]; ar[5] = y2[1]; ar[6] = y2[2]; ar[7] = y2[3];
}

// ---------------------------------------------------------------------------
// K2: depth-3 streamed signature of a (C=8). One 256-thread WG per batch.
// Thread T owns s2[j,k] and s3[j,k,l0],s3[j,k,l0+1] with
//   j=T>>5, k=(T>>2)&7, l0=2*(T&3).  s1(t) == a(t) (basepoint 0).
// ---------------------------------------------------------------------------
__global__ __launch_bounds__(256) void sig1_scan_kernel(
    const float* __restrict__ a, float* __restrict__ bs) {
  const int batch = blockIdx.x;
  const int T = threadIdx.x;
  const int j  = T >> 5;
  const int k  = (T >> 2) & 7;
  const int l0 = (T & 3) * 2;
  const float* arow = a  + (size_t)batch * L1 * C1;
  float*       brow = bs + (size_t)batch * L1 * SIG1;

  float s1j = 0.f, s2v = 0.f, s30 = 0.f, s31 = 0.f;
  float pj = 0.f, pk = 0.f, pl0 = 0.f, pl1 = 0.f;

  for (int t = 0; t < L1; ++t) {
    const float* r = arow + t * C1;
    float  cj = r[j];
    float  ck = r[k];
    float2 cl = *(const float2*)(r + l0);
    float dj = cj - pj, dk = ck - pk, dl0 = cl.x - pl0, dl1 = cl.y - pl1;
    pj = cj; pk = ck; pl0 = cl.x; pl1 = cl.y;

    float s2o = s2v;
    float djk = dj * dk;
    s30 += s2o * dl0 + 0.5f * s1j * dk * dl0 + djk * dl0 * (1.f / 6.f);
    s31 += s2o * dl1 + 0.5f * s1j * dk * dl1 + djk * dl1 * (1.f / 6.f);
    s2v += s1j * dk + 0.5f * djk;
    s1j += dj;

    float* o = brow + (size_t)t * SIG1;
    if (T < 8) o[T] = r[T];                    // n1 == prefix sum == a(t)
    if ((T & 3) == 0) o[8 + (T >> 2)] = s2v;   // n2 flat (j*8+k) == T>>2
    float2 st; st.x = s30; st.y = s31;
    *(float2*)(o + 72 + 2 * T) = st;           // n3 flat base 8+64
  }
}

// ---------------------------------------------------------------------------
// K3: augment2 via f32 WMMA 16x16x4. One wave per 16-position tile.
// GEMM: M=16 positions, N=16 (8 valid out ch), K=2336 (tap*584+ch).
// A lane layout (16x4 f32): lanes 0-15 K{0,1}, lanes 16-31 K{2,3}.
// Then 8->8->4 1x1 tail through per-wave LDS (same-wave DS is in-order).
// ---------------------------------------------------------------------------
__global__ __launch_bounds__(256) void conv2_wmma_kernel(
    const float* __restrict__ bs,
    const float* __restrict__ w0, const float* __restrict__ b0,
    const float* __restrict__ w1, const float* __restrict__ b1,
    const float* __restrict__ w2, const float* __restrict__ b2,
    float* __restrict__ cout) {
  __shared__ float hbuf[8][16][8];
  const int tid  = threadIdx.x;
  const int wid  = tid >> 5;
  const int lane = tid & 31;
  const int tile = blockIdx.x * 8 + wid;         // 4096 tiles total
  const int batch = tile >> 6;
  const int pos0  = (tile & 63) * 16;
  const int m  = lane & 15;                      // A row / B col
  const int kh = lane >> 4;                      // 0: K{0,1}, 1: K{2,3}
  int p = pos0 + m; if (p > L2 - 1) p = L2 - 1;  // clamp (stores guarded)
  const float wmsk = (m < 8) ? 1.f : 0.f;        // cols 8..15 are zero-pad

  v8f acc = {0.f, 0.f, 0.f, 0.f, 0.f, 0.f, 0.f, 0.f};
  for (int tap = 0; tap < 4; ++tap) {
    const float* ar = bs + ((size_t)batch * L1 + p + tap) * SIG1 + kh * 2;
    const float* wr = w0 + (size_t)tap * SIG1 * 8 + (kh * 2) * 8 + (m & 7);
    for (int cg = 0; cg < 146; ++cg) {           // 584 ch / 4 per WMMA
      float2 av = *(const float2*)(ar + cg * 4);
      const float* wp = wr + cg * 32;            // (ch0)*8 step
      v2f af; af.x = av.x;          af.y = av.y;
      v2f bf; bf.x = wp[0] * wmsk;  bf.y = wp[8] * wmsk;
      acc = __builtin_amdgcn_wmma_f32_16x16x4_f32(
          false, af, false, bf, (short)0, acc, false, false);
    }
  }

  // bias + ReLU; C layout: VGPR r -> M=r (lanes 0-15) / M=8+r (lanes 16-31)
  const float bias = (m < 8) ? b0[m] : 0.f;
#pragma unroll
  for (int r2 = 0; r2 < 8; ++r2) {
    float v = acc[r2] + bias;
    acc[r2] = v > 0.f ? v : 0.f;
  }
  if (m < 8) {
#pragma unroll
    for (int r2 = 0; r2 < 8; ++r2) hbuf[wid][r2 + kh * 8][m] = acc[r2];
  }
  __builtin_amdgcn_wave_barrier();               // same-wave DS is in-order

  if (lane < 16) {
    int p2 = pos0 + lane;
    if (p2 < L2) {
      float h[8];
#pragma unroll
      for (int c = 0; c < 8; ++c) h[c] = hbuf[wid][lane][c];
      float y1[8];
#pragma unroll
      for (int o = 0; o < 8; ++o) {
        float s = b1[o];
#pragma unroll
        for (int c = 0; c < 8; ++c) s += h[c] * w1[c * 8 + o];
        y1[o] = s > 0.f ? s : 0.f;
      }
      float yo[4];
#pragma unroll
      for (int o = 0; o < 4; ++o) {
        float s = b2[o];
#pragma unroll
        for (int c = 0; c < 8; ++c) s += y1[c] * w2[c * 4 + o];
        yo[o] = s;
      }
      float4 v; v.x = yo[0]; v.y = yo[1]; v.z = yo[2]; v.w = yo[3];
      *(float4*)(cout + ((size_t)batch * L2 + p2) * 4) = v;
    }
  }
}

// ---------------------------------------------------------------------------
// K4: final depth-3 signature (C=4) + linear head. One wave per batch.
// Lane owns s2[j,k] and s3 pair; s1(final) == c[last]. LDS-staged chunks.
// ---------------------------------------------------------------------------
__global__ __launch_bounds__(32) void sig2_scan_kernel(
    const float* __restrict__ cin,
    const float* __restrict__ lw, const float* __restrict__ lb,
    float* __restrict__ out) {
  __shared__ float buf[256 * 4];
  const int batch = blockIdx.x;
  const int lane  = threadIdx.x;
  const int j  = lane >> 3;
  const int k  = (lane >> 1) & 3;
  const int l0 = (lane & 1) * 2;
  const float* crow = cin + (size_t)batch * L2 * 4;

  float s1j = 0.f, s2v = 0.f, s30 = 0.f, s31 = 0.f;
  float pj = 0.f, pk = 0.f, pl0 = 0.f, pl1 = 0.f;

  for (int cb = 0; cb < L2; cb += 256) {
    const int cnt = (L2 - cb) < 256 ? (L2 - cb) : 256;
    __syncthreads();
#pragma unroll
    for (int r = 0; r < 8; ++r) {
      int s = cb + r * 32 + lane;
      if (s < L2)
        *(float4*)(buf + (r * 32 + lane) * 4) = *(const float4*)(crow + (size_t)s * 4);
    }
    __syncthreads();
    for (int s = 0; s < cnt; ++s) {
      const float* r = buf + s * 4;
      float  cj = r[j];
      float  ck = r[k];
      float2 cl = *(const float2*)(r + l0);
      float dj = cj - pj, dk = ck - pk, dl0 = cl.x - pl0, dl1 = cl.y - pl1;
      pj = cj; pk = ck; pl0 = cl.x; pl1 = cl.y;
      float s2o = s2v;
      float djk = dj * dk;
      s30 += s2o * dl0 + 0.5f * s1j * dk * dl0 + djk * dl0 * (1.f / 6.f);
      s31 += s2o * dl1 + 0.5f * s1j * dk * dl1 + djk * dl1 * (1.f / 6.f);
      s2v += s1j * dk + 0.5f * djk;
      s1j += dj;
    }
  }

  // fused linear head: sig flat = [s1(0-3), s2(4-19), s3(20-83)]
  float a0 = s30 * lw[(20 + 2 * lane) * 2 + 0] + s31 * lw[(21 + 2 * lane) * 2 + 0];
  float a1 = s30 * lw[(20 + 2 * lane) * 2 + 1] + s31 * lw[(21 + 2 * lane) * 2 + 1];
  if ((lane & 1) == 0) {
    int u = 4 + (lane >> 1);                      // unique s2 flat index
    a0 += s2v * lw[u * 2 + 0];
    a1 += s2v * lw[u * 2 + 1];
  }
  if (lane == 0) {                                // s1(final) == c[last]
    float4 fin = *(const float4*)(crow + (size_t)(L2 - 1) * 4);
    a0 += fin.x * lw[0] + fin.y * lw[2] + fin.z * lw[4] + fin.w * lw[6];
    a1 += fin.x * lw[1] + fin.y * lw[3] + fin.z * lw[5] + fin.w * lw[7];
  }
#pragma unroll
  for (int off = 16; off > 0; off >>= 1) {
    a0 += __shfl_xor(a0, off, 32);
    a1 += __shfl_xor(a1, off, 32);
  }
  if (lane == 0) {
    out[batch * 2 + 0] = a0 + lb[0];
    out[batch * 2 + 1] = a1 + lb[1];
  }
}

// ---------------------------------------------------------------------------
extern "C" void kernel_launch(void* const* d_in, const int* in_sizes, int n_in,
                              void* d_out, int out_size, void* d_ws, size_t ws_size,
                              hipStream_t stream) {
  (void)in_sizes; (void)n_in; (void)out_size; (void)ws_size;
  const float* x     = (const float*)d_in[0];
  const float* a1w0  = (const float*)d_in[1];
  const float* a1b0  = (const float*)d_in[2];
  const float* a1w1  = (const float*)d_in[3];
  const float* a1b1  = (const float*)d_in[4];
  const float* a1w2  = (const float*)d_in[5];
  const float* a1b2  = (const float*)d_in[6];
  const float* a2w0  = (const float*)d_in[7];
  const float* a2b0  = (const float*)d_in[8];
  const float* a2w1  = (const float*)d_in[9];
  const float* a2b1  = (const float*)d_in[10];
  const float* a2w2  = (const float*)d_in[11];
  const float* a2b2  = (const float*)d_in[12];
  const float* lin_w = (const float*)d_in[13];
  const float* lin_b = (const float*)d_in[14];

  float* ws = (float*)d_ws;
  float* a  = ws;                  // (64,1021,8)
  float* bs = ws + A_SZ;           // (64,1021,584) — fits MI455X 192MB L2
  float* c  = ws + A_SZ + S_SZ;    // (64,1018,4)
  float* out = (float*)d_out;      // (64,2)

  augment1_kernel<<<256, 256, 0, stream>>>(x, a1w0, a1b0, a1w1, a1b1, a1w2, a1b2, a);
  sig1_scan_kernel<<<B_N, 256, 0, stream>>>(a, bs);
  conv2_wmma_kernel<<<512, 256, 0, stream>>>(bs, a2w0, a2b0, a2w1, a2b1, a2w2, a2b2, c);
  sig2_scan_kernel<<<B_N, 32, 0, stream>>>(c, lin_w, lin_b, out);
}